// GAT_4887672783343
// MI455X (gfx1250) — compile-verified
//
#include <hip/hip_runtime.h>
#include <cstdint>
#include <cstddef>

typedef __attribute__((ext_vector_type(2))) float v2f;
typedef __attribute__((ext_vector_type(8))) float v8f;

#define GAT_NEG_SLOPE 0.2f

// Monotone order-preserving float->uint map so segment-max can use atomicMax(u32).
__device__ __forceinline__ unsigned fflip(float f) {
  unsigned u = __float_as_uint(f);
  return (u & 0x80000000u) ? ~u : (u | 0x80000000u);
}
__device__ __forceinline__ float funflip(unsigned u) {
  return __uint_as_float((u & 0x80000000u) ? (u & 0x7fffffffu) : ~u);
}
__device__ __forceinline__ float lrelu(float v) { return v > 0.0f ? v : GAT_NEG_SLOPE * v; }

// ---------------------------------------------------------------------------
// h[N,64] = x[N,64] @ W[64,64] via V_WMMA_F32_16X16X4_F32.
// One wave per 16x16 output tile; block = 4 waves = 4 col tiles (full 64 cols).
// A layout: lane l -> row (l&15), k-pair 2*(l>>4).  B: lane l -> col (l&15),
// same k-pair.  C/D: VGPR i -> row i + 8*(l>>4), col (l&15).
// ---------------------------------------------------------------------------
__global__ void gat_gemm64x64(const float* __restrict__ x, const float* __restrict__ W,
                              float* __restrict__ h, int N) {
  const int lane = threadIdx.x & 31;
  const int wave = threadIdx.x >> 5;          // col tile 0..3
  const int row0 = blockIdx.x * 16;
  if (row0 >= N) return;
  const int col0 = wave * 16;
  const int r  = lane & 15;
  const int kh = lane >> 4;                   // which k-pair (0 or 1)
  int rr = row0 + r; if (rr >= N) rr = N - 1; // clamp (N%16==0 in practice)
  const float* xrow = x + (size_t)rr * 64;
  v8f acc = {};
#pragma unroll
  for (int kk = 0; kk < 64; kk += 4) {
    const int k0 = kk + 2 * kh;
    v2f a, b;
    a.x = xrow[k0];
    a.y = xrow[k0 + 1];
    b.x = W[(size_t)k0 * 64 + col0 + r];
    b.y = W[(size_t)(k0 + 1) * 64 + col0 + r];
    acc = __builtin_amdgcn_wmma_f32_16x16x4_f32(false, a, false, b, (short)0, acc, false, false);
  }
#pragma unroll
  for (int i = 0; i < 8; ++i) {
    const int orow = row0 + i + kh * 8;
    if (orow < N) h[(size_t)orow * 64 + col0 + r] = acc[i];
  }
}

// h2[N,8] = h[N,64] @ W2[64,8]; B columns padded to 16 with zeros.
__global__ void gat_gemm64x8(const float* __restrict__ h, const float* __restrict__ W2,
                             float* __restrict__ h2, int N) {
  const int lane = threadIdx.x & 31;
  const int wave = threadIdx.x >> 5;
  const int row0 = (blockIdx.x * 4 + wave) * 16;
  if (row0 >= N) return;
  const int r  = lane & 15;
  const int kh = lane >> 4;
  int rr = row0 + r; if (rr >= N) rr = N - 1;
  const float* hrow = h + (size_t)rr * 64;
  v8f acc = {};
#pragma unroll
  for (int kk = 0; kk < 64; kk += 4) {
    const int k0 = kk + 2 * kh;
    v2f a, b;
    a.x = hrow[k0];
    a.y = hrow[k0 + 1];
    b.x = (r < 8) ? W2[(size_t)k0 * 8 + r] : 0.0f;
    b.y = (r < 8) ? W2[(size_t)(k0 + 1) * 8 + r] : 0.0f;
    acc = __builtin_amdgcn_wmma_f32_16x16x4_f32(false, a, false, b, (short)0, acc, false, false);
  }
  if (r < 8) {
#pragma unroll
    for (int i = 0; i < 8; ++i) {
      const int orow = row0 + i + kh * 8;
      if (orow < N) h2[(size_t)orow * 8 + r] = acc[i];
    }
  }
}

// asrc[n,h] = <h[n,h,:], a_src[h,:]>, adst likewise (C=8 per head).
template <int H>
__global__ void gat_attn_coef(const float* __restrict__ h, const float* __restrict__ a_src,
                              const float* __restrict__ a_dst, float* __restrict__ asrc,
                              float* __restrict__ adst, int N) {
  const int tid = blockIdx.x * blockDim.x + threadIdx.x;
  if (tid >= N * H) return;
  const int n = tid / H, hh = tid % H;
  const float* hp = h + (size_t)n * (H * 8) + hh * 8;
  const float* as = a_src + hh * 8;
  const float* ad = a_dst + hh * 8;
  float ss = 0.0f, dd = 0.0f;
#pragma unroll
  for (int c = 0; c < 8; ++c) { ss += hp[c] * as[c]; dd += hp[c] * ad[c]; }
  asrc[tid] = ss;
  adst[tid] = dd;
}

// Decode (edge, head) -> (src, dst); indices e >= E are self-loops (node e-E).
template <int H>
__device__ __forceinline__ void edge_sd(const long long* __restrict__ src,
                                        const long long* __restrict__ dst,
                                        int e, int E, int& s, int& d) {
  if (e < E) { s = (int)src[e]; d = (int)dst[e]; }
  else       { s = d = e - E; }
}

template <int H>
__global__ void gat_edge_max(const long long* __restrict__ src, const long long* __restrict__ dst,
                             const float* __restrict__ asrc, const float* __restrict__ adst,
                             unsigned* __restrict__ mflip, int E, int N) {
  const int tid = blockIdx.x * blockDim.x + threadIdx.x;
  if (tid >= (E + N) * H) return;
  const int hh = tid % H, e = tid / H;
  int s, d; edge_sd<H>(src, dst, e, E, s, d);
  const float ev = lrelu(asrc[s * H + hh] + adst[d * H + hh]);
  atomicMax(&mflip[d * H + hh], fflip(ev));
}

template <int H>
__global__ void gat_edge_sum(const long long* __restrict__ src, const long long* __restrict__ dst,
                             const float* __restrict__ asrc, const float* __restrict__ adst,
                             const unsigned* __restrict__ mflip, float* __restrict__ ssum,
                             int E, int N) {
  const int tid = blockIdx.x * blockDim.x + threadIdx.x;
  if (tid >= (E + N) * H) return;
  const int hh = tid % H, e = tid / H;
  int s, d; edge_sd<H>(src, dst, e, E, s, d);
  const float ev = lrelu(asrc[s * H + hh] + adst[d * H + hh]);
  const float m = funflip(mflip[d * H + hh]);
  atomicAdd(&ssum[d * H + hh], expf(ev - m));
}

template <int H>
__global__ void gat_edge_agg(const long long* __restrict__ src, const long long* __restrict__ dst,
                             const float* __restrict__ asrc, const float* __restrict__ adst,
                             const unsigned* __restrict__ mflip, const float* __restrict__ ssum,
                             const float* __restrict__ h, float* __restrict__ outp,
                             int E, int N) {
  const int tid = blockIdx.x * blockDim.x + threadIdx.x;
  if (tid >= (E + N) * H) return;
  const int hh = tid % H, e = tid / H;
  int s, d; edge_sd<H>(src, dst, e, E, s, d);
  const int di = d * H + hh;
  const float ev = lrelu(asrc[s * H + hh] + adst[di]);
  const float alpha = expf(ev - funflip(mflip[di])) / (ssum[di] + 1e-16f);
  const float* hs = h + (size_t)s * (H * 8) + hh * 8;
  float* o = outp + (size_t)d * (H * 8) + hh * 8;
#pragma unroll
  for (int c = 0; c < 8; ++c) atomicAdd(&o[c], alpha * hs[c]);
}

// out[n,j] = elu(agg[n,j] + b[j])   (F = 64, layer-1 concat output + bias + ELU)
__global__ void gat_bias_elu(const float* __restrict__ agg, const float* __restrict__ b,
                             float* __restrict__ outp, int N, int F) {
  const int tid = blockIdx.x * blockDim.x + threadIdx.x;
  if (tid >= N * F) return;
  const float v = agg[tid] + b[tid % F];
  outp[tid] = v > 0.0f ? v : expm1f(v);
}

// out[n,o] = sum_c (agg2[n,c] + b2[c]) * Wh[c,o] + bh[o]   (8 -> 16)
__global__ void gat_head(const float* __restrict__ agg2, const float* __restrict__ b2,
                         const float* __restrict__ Wh, const float* __restrict__ bh,
                         float* __restrict__ outp, int N) {
  const int tid = blockIdx.x * blockDim.x + threadIdx.x;
  if (tid >= N * 16) return;
  const int n = tid >> 4, o = tid & 15;
  float s = bh[o];
#pragma unroll
  for (int c = 0; c < 8; ++c) s += (agg2[(size_t)n * 8 + c] + b2[c]) * Wh[c * 16 + o];
  outp[tid] = s;
}

extern "C" void kernel_launch(void* const* d_in, const int* in_sizes, int n_in,
                              void* d_out, int out_size, void* d_ws, size_t ws_size,
                              hipStream_t stream) {
  const float*      x      = (const float*)d_in[0];
  const long long*  ei     = (const long long*)d_in[1];   // int64 [2,E]
  const float*      W1     = (const float*)d_in[2];
  const float*      a_src1 = (const float*)d_in[3];
  const float*      a_dst1 = (const float*)d_in[4];
  const float*      b1     = (const float*)d_in[5];
  const float*      W2     = (const float*)d_in[6];
  const float*      a_src2 = (const float*)d_in[7];
  const float*      a_dst2 = (const float*)d_in[8];
  const float*      b2     = (const float*)d_in[9];
  const float*      Wh     = (const float*)d_in[10];
  const float*      bh     = (const float*)d_in[11];
  float*            outp   = (float*)d_out;

  const int N = in_sizes[0] / 64;
  const int E = in_sizes[1] / 2;
  const long long* srcIdx = ei;
  const long long* dstIdx = ei + E;

  // Workspace layout (all 4-byte elements); zeroed region is contiguous.
  const size_t Ns = (size_t)N;
  float* ws    = (float*)d_ws;
  float* h1    = ws;                        // N*64 (reused as ELU output)
  float* asrc1 = h1 + Ns * 64;              // N*8
  float* adst1 = asrc1 + Ns * 8;            // N*8
  float* h2    = adst1 + Ns * 8;            // N*8
  float* asrc2 = h2 + Ns * 8;               // N
  float* adst2 = asrc2 + Ns;                // N
  float* zb    = adst2 + Ns;                // ---- zeroed region start ----
  unsigned* mflip1 = (unsigned*)zb;         // N*8   (0 < fflip(-inf): below-all sentinel)
  float* ssum1 = zb + Ns * 8;               // N*8
  float* out1  = ssum1 + Ns * 8;            // N*64
  unsigned* mflip2 = (unsigned*)(out1 + Ns * 64); // N
  float* ssum2 = out1 + Ns * 64 + Ns;       // N
  float* out2  = ssum2 + Ns;                // N*8
  hipMemsetAsync(zb, 0, Ns * 90 * sizeof(float), stream);

  const int rowTiles = (N + 15) / 16;
  int t;

  // ---- layer 1 (H=8, C=8) ----
  gat_gemm64x64<<<rowTiles, 128, 0, stream>>>(x, W1, h1, N);
  t = N * 8;
  gat_attn_coef<8><<<(t + 255) / 256, 256, 0, stream>>>(h1, a_src1, a_dst1, asrc1, adst1, N);
  t = (E + N) * 8;
  gat_edge_max<8><<<(t + 255) / 256, 256, 0, stream>>>(srcIdx, dstIdx, asrc1, adst1, mflip1, E, N);
  gat_edge_sum<8><<<(t + 255) / 256, 256, 0, stream>>>(srcIdx, dstIdx, asrc1, adst1, mflip1, ssum1, E, N);
  gat_edge_agg<8><<<(t + 255) / 256, 256, 0, stream>>>(srcIdx, dstIdx, asrc1, adst1, mflip1, ssum1,
                                                       h1, out1, E, N);
  t = N * 64;
  gat_bias_elu<<<(t + 255) / 256, 256, 0, stream>>>(out1, b1, h1, N, 64); // h1 := elu(out1+b1)

  // ---- layer 2 (H=1, C=8) ----
  gat_gemm64x8<<<(rowTiles + 3) / 4, 128, 0, stream>>>(h1, W2, h2, N);
  t = N;
  gat_attn_coef<1><<<(t + 255) / 256, 256, 0, stream>>>(h2, a_src2, a_dst2, asrc2, adst2, N);
  t = E + N;
  gat_edge_max<1><<<(t + 255) / 256, 256, 0, stream>>>(srcIdx, dstIdx, asrc2, adst2, mflip2, E, N);
  gat_edge_sum<1><<<(t + 255) / 256, 256, 0, stream>>>(srcIdx, dstIdx, asrc2, adst2, mflip2, ssum2, E, N);
  gat_edge_agg<1><<<(t + 255) / 256, 256, 0, stream>>>(srcIdx, dstIdx, asrc2, adst2, mflip2, ssum2,
                                                       h2, out2, E, N);

  // ---- post-MP head: (agg2 + b2) @ Wh + bh ----
  t = N * 16;
  gat_head<<<(t + 255) / 256, 256, 0, stream>>>(out2, b2, Wh, bh, outp, N);
}